// VolumeRenderingNeus_51780125721343
// MI455X (gfx1250) — compile-verified
//
#include <hip/hip_runtime.h>
#include <cstdint>

// ---------------------------------------------------------------------------
// NeuS volume rendering, MI455X (gfx1250, wave32).
// Bandwidth-bound (~220 MB traffic -> ~10us floor @ 23.3 TB/s). Single fused
// pass over samples; one wave32 per ray; in-register segmented scan.
// Streams are marked non-temporal (CDNA5 TH=NT) since every byte is touched
// exactly once; keeps L2 free for ray_start / per-ray outputs.
// ---------------------------------------------------------------------------

__device__ __forceinline__ float sigmoid_fast(float x) {
    return 1.0f / (1.0f + __expf(-x));
}

// Kernel 0: init ray_start[] to N (lower-bound default for empty trailing
// rays and sentinel ray_start[R] = N), and emit the inv_s scalar output.
__global__ void k0_init(int* __restrict__ ray_start, int R, int N,
                        const float* __restrict__ variance,
                        float* __restrict__ inv_s_out) {
    const int r = blockIdx.x * blockDim.x + threadIdx.x;
    if (r <= R) ray_start[r] = N;
    if (r == 0) {
        float is = __expf(variance[0] * 10.0f);
        is = fminf(fmaxf(is, 1e-6f), 1e6f);
        *inv_s_out = is;
    }
}

// Kernel 1: ray_start[r] = first sample index i with ray_ids[i] >= r.
// ray_ids is sorted, so each boundary thread fills the (prev, cur] range.
// Stages ray_ids through LDS with the CDNA5 async global->LDS path.
__global__ void __launch_bounds__(256)
k1_bounds(const int* __restrict__ ray_ids, int* __restrict__ ray_start, int N) {
    __shared__ int rid[257];   // rid[k] = ray_ids[base - 1 + k]
    const int tid  = threadIdx.x;
    const int base = blockIdx.x * 256;

    {
        const int g = base - 1 + tid;
        if (g >= 0 && g < N) {
            uint32_t lds = (uint32_t)(uintptr_t)&rid[tid];
            uint64_t ga  = (uint64_t)(uintptr_t)(ray_ids + g);
            asm volatile("global_load_async_to_lds_b32 %0, %1, off"
                         :: "v"(lds), "v"(ga) : "memory");
        }
        const int g2 = base + 255;
        if (tid == 0 && g2 < N) {
            uint32_t lds = (uint32_t)(uintptr_t)&rid[256];
            uint64_t ga  = (uint64_t)(uintptr_t)(ray_ids + g2);
            asm volatile("global_load_async_to_lds_b32 %0, %1, off"
                         :: "v"(lds), "v"(ga) : "memory");
        }
    }
    asm volatile("s_wait_asynccnt 0" ::: "memory");
    __syncthreads();

    const int i = base + tid;
    if (i >= N) return;
    const int cur = rid[tid + 1];          // ray_ids[i]
    if (i == 0) {
        for (int r = 0; r <= cur; ++r) ray_start[r] = 0;
    } else {
        const int prev = rid[tid];         // ray_ids[i-1]
        if (cur != prev) {
            for (int r = prev + 1; r <= cur; ++r) ray_start[r] = i;
        }
    }
}

// Kernel 2: one wave32 per ray. Segmented exclusive scan of log(1-alpha+1e-7)
// entirely in registers (rays are contiguous), fused with alpha computation
// and per-ray reductions. No atomics, fully deterministic.
__global__ void __launch_bounds__(256)
k2_render(const float* __restrict__ variance,
          const float* __restrict__ sdf,
          const float* __restrict__ grad,
          const float* __restrict__ dirs,
          const float* __restrict__ dt,
          const float* __restrict__ rgb,
          const float* __restrict__ cosann,
          const int*   __restrict__ ray_start,
          float* __restrict__ pred_rgb,
          float* __restrict__ weights,
          float* __restrict__ wsum_o,
          float* __restrict__ bg_o,
          int R) {
    const int wid  = (int)((blockIdx.x * (unsigned)blockDim.x + threadIdx.x) >> 5);
    const int lane = threadIdx.x & 31;
    if (wid >= R) return;

    const int s0 = ray_start[wid];
    const int s1 = ray_start[wid + 1];

    float inv_s = __expf(variance[0] * 10.0f);
    inv_s = fminf(fmaxf(inv_s, 1e-6f), 1e6f);
    const float ca = cosann[0];

    float run = 0.0f;                 // per-ray running sum of log(1-alpha)
    float wsum = 0.0f, a0 = 0.0f, a1 = 0.0f, a2 = 0.0f;

    for (int bse = s0; bse < s1; bse += 32) {
        const int  i   = bse + lane;
        const bool act = (i < s1);

        float sv = 0.0f, dtv = 0.0f;
        float g0 = 0.0f, g1 = 0.0f, g2 = 0.0f;
        float d0 = 0.0f, d1 = 0.0f, d2 = 0.0f;
        float c0 = 0.0f, c1 = 0.0f, c2 = 0.0f;
        if (act) {
            sv  = __builtin_nontemporal_load(sdf + i);
            dtv = __builtin_nontemporal_load(dt + i);
            const float* gp = grad + 3 * (size_t)i;
            g0 = __builtin_nontemporal_load(gp + 0);
            g1 = __builtin_nontemporal_load(gp + 1);
            g2 = __builtin_nontemporal_load(gp + 2);
            const float* dp = dirs + 3 * (size_t)i;
            d0 = __builtin_nontemporal_load(dp + 0);
            d1 = __builtin_nontemporal_load(dp + 1);
            d2 = __builtin_nontemporal_load(dp + 2);
            const float* cp = rgb + 3 * (size_t)i;
            c0 = __builtin_nontemporal_load(cp + 0);
            c1 = __builtin_nontemporal_load(cp + 1);
            c2 = __builtin_nontemporal_load(cp + 2);
        }
        // Prefetch next chunk of the wide streams (global_prefetch_b8).
        {
            const int j = i + 32;
            if (j < s1) {
                __builtin_prefetch(grad + 3 * (size_t)j, 0, 0);
                __builtin_prefetch(dirs + 3 * (size_t)j, 0, 0);
                __builtin_prefetch(rgb  + 3 * (size_t)j, 0, 0);
            }
        }

        const float tc = d0 * g0 + d1 * g1 + d2 * g2;
        const float ic = -(fmaxf(-tc * 0.5f + 0.5f, 0.0f) * (1.0f - ca)
                           + fmaxf(-tc, 0.0f) * ca);
        const float e  = ic * dtv * 0.5f;
        const float pc = sigmoid_fast((sv - e) * inv_s);
        const float nc = sigmoid_fast((sv + e) * inv_s);
        float alpha = (pc - nc + 1e-5f) / (pc + 1e-5f);
        alpha = fminf(fmaxf(alpha, 0.0f), 1.0f);

        const float logc = act ? __logf(1.0f - alpha + 1e-7f) : 0.0f;

        // wave32 inclusive scan of logc
        float incl = logc;
#pragma unroll
        for (int off = 1; off < 32; off <<= 1) {
            const float t = __shfl_up(incl, off, 32);
            if (lane >= off) incl += t;
        }

        const float T = __expf(run + (incl - logc));  // exclusive within ray
        const float w = alpha * T;
        if (act) {
            __builtin_nontemporal_store(w, weights + i);
            wsum += w;
            a0 += c0 * w; a1 += c1 * w; a2 += c2 * w;
        }
        run += __shfl(incl, 31, 32);  // carry chunk total across iterations
    }

    // wave reduction of the per-lane partial sums
#pragma unroll
    for (int off = 16; off >= 1; off >>= 1) {
        wsum += __shfl_xor(wsum, off, 32);
        a0   += __shfl_xor(a0,   off, 32);
        a1   += __shfl_xor(a1,   off, 32);
        a2   += __shfl_xor(a2,   off, 32);
    }
    if (lane == 0) {
        wsum_o[wid] = wsum;
        bg_o[wid]   = __expf(run);   // empty ray -> exp(0) = 1, matches ref
        float* pr = pred_rgb + 3 * (size_t)wid;
        pr[0] = a0; pr[1] = a1; pr[2] = a2;
    }
}

extern "C" void kernel_launch(void* const* d_in, const int* in_sizes, int n_in,
                              void* d_out, int out_size, void* d_ws, size_t ws_size,
                              hipStream_t stream) {
    const float* variance = (const float*)d_in[0];
    const float* sdf      = (const float*)d_in[1];
    const float* grad     = (const float*)d_in[2];
    const float* dirs     = (const float*)d_in[3];
    const float* dt       = (const float*)d_in[4];
    const float* rgb      = (const float*)d_in[5];
    const float* cosann   = (const float*)d_in[6];
    const int*   ray_ids  = (const int*)d_in[7];

    const int N = in_sizes[1];                 // N_SAMPLES
    const int R = (out_size - 1 - N) / 5;      // out = 3R + N + R + R + 1

    float* out       = (float*)d_out;
    float* pred_rgb  = out;                                // [R,3]
    float* weights   = out + (size_t)3 * R;                // [N,1]
    float* wsum      = out + (size_t)3 * R + N;            // [R,1]
    float* bg        = out + (size_t)4 * R + N;            // [R,1]
    float* inv_s_out = out + (size_t)5 * R + N;            // [1]

    int* ray_start = (int*)d_ws;                           // (R+1) ints

    k0_init<<<(R + 256) / 256, 256, 0, stream>>>(ray_start, R, N, variance, inv_s_out);
    k1_bounds<<<(N + 255) / 256, 256, 0, stream>>>(ray_ids, ray_start, N);

    const long long total = (long long)R * 32;
    k2_render<<<(int)((total + 255) / 256), 256, 0, stream>>>(
        variance, sdf, grad, dirs, dt, rgb, cosann,
        ray_start, pred_rgb, weights, wsum, bg, R);
}